// CustomLoss_74826920231413
// MI455X (gfx1250) — compile-verified
//
#include <hip/hip_runtime.h>
#include <hip/hip_bf16.h>

typedef __attribute__((ext_vector_type(2))) float v2f;
typedef __attribute__((ext_vector_type(8))) float v8f;

#define E_DIM 1024
#define K_DIM 20
#define NOUT_C 40980   // K*(2E+1)
#define VLOC_C 20500   // K + K*E
#define NPAIR 190      // K*(K-1)/2
#define PENALTY_C 0.01f

// ---------------------------------------------------------------------------
// ws layout (floats): [0..3] = {num, den, objU, objV}; then 4 arrays of B*K*E:
// ArT, AiT, VrT, ViT stored [b][k][e] (k-major rows of length E).
// ---------------------------------------------------------------------------

__global__ void zero_acc_kernel(float* acc) {
    if (threadIdx.x < 4) acc[threadIdx.x] = 0.0f;
}

// Build A = U*d (complex) and transposed copies of V, laid out [b][k][e].
__global__ void pack_kernel(const float* __restrict__ nn,
                            float* __restrict__ ArT, float* __restrict__ AiT,
                            float* __restrict__ VrT, float* __restrict__ ViT,
                            int B) {
    int idx = blockIdx.x * blockDim.x + threadIdx.x;   // over B*K*E, e fastest
    int total = B * K_DIM * E_DIM;
    if (idx >= total) return;
    int e = idx % E_DIM;
    int k = (idx / E_DIM) % K_DIM;
    int b = idx / (E_DIM * K_DIM);

    const float* row = nn + (size_t)b * (2 * NOUT_C);
    float ur = row[K_DIM + e * K_DIM + k];
    float ui = row[NOUT_C + K_DIM + e * K_DIM + k];
    float dr = row[k];
    float di = row[NOUT_C + k];
    float vr = row[VLOC_C + e * K_DIM + k];
    float vi = row[NOUT_C + VLOC_C + e * K_DIM + k];

    size_t o = ((size_t)b * K_DIM + k) * E_DIM + e;
    ArT[o] = ur * dr - ui * di;
    AiT[o] = ur * di + ui * dr;
    VrT[o] = vr;
    ViT[o] = vi;
}

// Fused batched complex GEMM (rank-20) + residual/norm reduction.
// One wave32 per 16x16 (e,f) tile; pred never touches memory.
__global__ void __launch_bounds__(256)
gemm_reduce_kernel(const float* __restrict__ ArT, const float* __restrict__ AiT,
                   const float* __restrict__ VrT, const float* __restrict__ ViT,
                   const float* __restrict__ kern_r, const float* __restrict__ kern_i,
                   float* __restrict__ acc) {
    const int b    = blockIdx.y;
    const int wave = threadIdx.x >> 5;
    const int lane = threadIdx.x & 31;
    const int row  = lane & 15;
    const int hi   = lane >> 4;                 // 0: lanes 0-15, 1: lanes 16-31

    const int tile = blockIdx.x * 8 + wave;     // 0 .. 64*64-1
    const int e0   = (tile >> 6) * 16;
    const int f0   = (tile & 63) * 16;

    const size_t abase = (size_t)b * K_DIM * E_DIM;
    const float* Ar = ArT + abase;
    const float* Ai = AiT + abase;
    const float* Vr = VrT + abase;
    const float* Vi = ViT + abase;

    // kern tile base for this lane: rows e0+8*hi+j (j=0..7), col f0+row
    const size_t kbase = ((size_t)b * E_DIM + e0 + 8 * hi) * E_DIM + f0 + row;
    const float* krp = kern_r + kbase;
    const float* kip = kern_i + kbase;
    // Prefetch the kern tile so HBM latency overlaps the WMMA chain.
    #pragma unroll
    for (int j = 0; j < 8; ++j) {
        __builtin_prefetch(&krp[(size_t)j * E_DIM], 0, 0);
        __builtin_prefetch(&kip[(size_t)j * E_DIM], 0, 0);
    }

    // Load A (16x4) and B (4x16) fragments for the 5 k-chunks.
    // ISA layout: VGPR0 holds K=k0 (lanes 0-15) / K=k0+2 (lanes 16-31); VGPR1 = +1.
    v2f ar[5], ai[5], br[5], bi[5];
    #pragma unroll
    for (int kc = 0; kc < 5; ++kc) {
        const int k0 = kc * 4 + 2 * hi;
        const size_t oa0 = (size_t)k0 * E_DIM + e0 + row;
        const size_t oa1 = (size_t)(k0 + 1) * E_DIM + e0 + row;
        const size_t ob0 = (size_t)k0 * E_DIM + f0 + row;
        const size_t ob1 = (size_t)(k0 + 1) * E_DIM + f0 + row;
        ar[kc][0] = Ar[oa0]; ar[kc][1] = Ar[oa1];
        ai[kc][0] = Ai[oa0]; ai[kc][1] = Ai[oa1];
        br[kc][0] = Vr[ob0]; br[kc][1] = Vr[ob1];
        bi[kc][0] = Vi[ob0]; bi[kc][1] = Vi[ob1];
    }

    v8f rr = {}, ii = {}, ri = {}, ir = {};
    #pragma unroll
    for (int kc = 0; kc < 5; ++kc) {
        rr = __builtin_amdgcn_wmma_f32_16x16x4_f32(false, ar[kc], false, br[kc],
                                                   (short)0, rr, false, false);
        ii = __builtin_amdgcn_wmma_f32_16x16x4_f32(false, ai[kc], false, bi[kc],
                                                   (short)0, ii, false, false);
        ri = __builtin_amdgcn_wmma_f32_16x16x4_f32(false, ar[kc], false, bi[kc],
                                                   (short)0, ri, false, false);
        ir = __builtin_amdgcn_wmma_f32_16x16x4_f32(false, ai[kc], false, br[kc],
                                                   (short)0, ir, false, false);
    }

    // C/D layout: VGPR j, lanes 0-15 -> M=j; lanes 16-31 -> M=j+8; N = lane&15.
    float num = 0.0f, den = 0.0f;
    #pragma unroll
    for (int j = 0; j < 8; ++j) {
        const float pr = rr[j] - ii[j];
        const float pi = ri[j] + ir[j];
        const float kr = krp[(size_t)j * E_DIM];
        const float ki = kip[(size_t)j * E_DIM];
        const float dr = kr - pr;
        const float di = ki - pi;
        num += dr * dr + di * di;
        den += kr * kr + ki * ki;
    }

    // Wave32 butterfly reduction, then one atomic per wave.
    #pragma unroll
    for (int off = 16; off >= 1; off >>= 1) {
        num += __shfl_xor(num, off, 32);
        den += __shfl_xor(den, off, 32);
    }
    if (lane == 0) {
        atomicAdd(&acc[0], num);
        atomicAdd(&acc[1], den);
    }
}

// Orthogonality: |(M^T M)[j,k]| summed over j<k, for M = U (y=0) or V (y=1).
__global__ void ortho_kernel(const float* __restrict__ nn, float* __restrict__ acc,
                             int B) {
    const int b = blockIdx.x / NPAIR;
    int p = blockIdx.x % NPAIR;
    int j = 0;
    while (p >= K_DIM - 1 - j) { p -= K_DIM - 1 - j; ++j; }
    const int k = j + 1 + p;
    const int base = (blockIdx.y == 0) ? K_DIM : VLOC_C;

    const float* rowp = nn + (size_t)b * (2 * NOUT_C);
    float gr = 0.0f, gi = 0.0f;
    for (int e = threadIdx.x; e < E_DIM; e += blockDim.x) {
        const int o = base + e * K_DIM;
        const float xr = rowp[o + j], xi = rowp[NOUT_C + o + j];
        const float yr = rowp[o + k], yi = rowp[NOUT_C + o + k];
        gr += xr * yr - xi * yi;
        gi += xr * yi + xi * yr;
    }
    __shared__ float sr[256], si[256];
    sr[threadIdx.x] = gr; si[threadIdx.x] = gi;
    __syncthreads();
    for (int st = 128; st > 0; st >>= 1) {
        if ((int)threadIdx.x < st) {
            sr[threadIdx.x] += sr[threadIdx.x + st];
            si[threadIdx.x] += si[threadIdx.x + st];
        }
        __syncthreads();
    }
    if (threadIdx.x == 0)
        atomicAdd(&acc[2 + blockIdx.y], sqrtf(sr[0] * sr[0] + si[0] * si[0]));
}

__global__ void finalize_kernel(const float* __restrict__ acc, float* __restrict__ out,
                                float invB) {
    if (threadIdx.x == 0 && blockIdx.x == 0) {
        const float obj1 = acc[2] * invB;
        const float obj2 = acc[3] * invB;
        out[0] = acc[0] / acc[1] + PENALTY_C * obj1 + PENALTY_C * obj2;
        out[1] = obj1;
        out[2] = obj2;
    }
}

extern "C" void kernel_launch(void* const* d_in, const int* in_sizes, int n_in,
                              void* d_out, int out_size, void* d_ws, size_t ws_size,
                              hipStream_t stream) {
    const float* nn     = (const float*)d_in[0];
    const float* kern_r = (const float*)d_in[1];
    const float* kern_i = (const float*)d_in[2];
    const int B = in_sizes[1] / (E_DIM * E_DIM);   // 32

    float* ws  = (float*)d_ws;
    float* acc = ws;                               // 4 accumulators (padded to 16)
    const size_t N1 = (size_t)B * K_DIM * E_DIM;   // 655360 floats per array
    float* ArT = ws + 16;
    float* AiT = ArT + N1;
    float* VrT = AiT + N1;
    float* ViT = VrT + N1;

    zero_acc_kernel<<<1, 32, 0, stream>>>(acc);

    const int total = B * K_DIM * E_DIM;
    pack_kernel<<<(total + 255) / 256, 256, 0, stream>>>(nn, ArT, AiT, VrT, ViT, B);

    // 64x64 tiles of 16x16 per batch, 8 waves (tiles) per 256-thread block.
    dim3 ggrid((E_DIM / 16) * (E_DIM / 16) / 8, B);
    gemm_reduce_kernel<<<ggrid, 256, 0, stream>>>(ArT, AiT, VrT, ViT,
                                                  kern_r, kern_i, acc);

    dim3 ogrid(B * NPAIR, 2);
    ortho_kernel<<<ogrid, 256, 0, stream>>>(nn, acc, B);

    finalize_kernel<<<1, 32, 0, stream>>>(acc, (float*)d_out, 1.0f / (float)B);
}